// DepthwiseTensorProduct_34299608825874
// MI455X (gfx1250) — compile-verified
//
#include <hip/hip_runtime.h>
#include <math.h>

// ---------------------------------------------------------------------------
// DepthwiseTensorProduct (e3nn uvu TP): per-edge  out = w ⊙ (A[64x9] @ G[9x51])
// G built per edge from x2[9] and constant (alpha-folded) Wigner-3j table W.
// Matmul mapped onto fp32 WMMA: V_WMMA_F32_16X16X4_F32, 48 per edge.
// Memory-bound: ~19.2 KB/edge -> ~41 us at 23.3 TB/s. Output staged through
// LDS so all global stores are coalesced 128-bit (full cacheline) writes.
// ---------------------------------------------------------------------------

typedef __attribute__((ext_vector_type(2))) float v2f;
typedef __attribute__((ext_vector_type(8))) float v8f;

// Sorted path metadata (order after e3nn Irreps.sort()):
// p : (l1,l2,l3) = (0,0,0)(1,1,0)(2,2,0)(0,1,1)(1,0,1)(1,2,1)(2,1,1)(1,1,1)
//                  (2,2,1)(0,2,2)(1,1,2)(2,0,2)(2,2,2)(1,2,2)(2,1,2)
__constant__ int c_L1[15]       = {0,1,2,0,1,1,2,1,2,0,1,2,2,1,2};
__constant__ int c_L2[15]       = {0,1,2,1,0,2,1,1,2,2,1,0,2,2,1};
__constant__ int c_L3[15]       = {0,0,0,1,1,1,1,1,1,2,2,2,2,2,2};
__constant__ int c_colStart[15] = {0,1,2,3,6,9,12,15,18,21,26,31,36,41,46};
__constant__ int c_outOff[15]   = {0,64,128,192,384,576,768,960,1152,1344,1664,1984,2304,2624,2944};
__constant__ int c_pdim[15]     = {1,1,1,3,3,3,3,3,3,5,5,5,5,5,5};
__constant__ int c_col2p[51] = {0,1,2, 3,3,3, 4,4,4, 5,5,5, 6,6,6, 7,7,7, 8,8,8,
                               9,9,9,9,9, 10,10,10,10,10, 11,11,11,11,11,
                               12,12,12,12,12, 13,13,13,13,13, 14,14,14,14,14};
__constant__ int c_col2k[51] = {0,0,0, 0,1,2, 0,1,2, 0,1,2, 0,1,2, 0,1,2, 0,1,2,
                               0,1,2,3,4, 0,1,2,3,4, 0,1,2,3,4,
                               0,1,2,3,4, 0,1,2,3,4, 0,1,2,3,4};

// ------------------------- init: Wigner 3j table ---------------------------

__device__ __forceinline__ double dfact(int n) {
  double r = 1.0;
  for (int i = 2; i <= n; ++i) r *= (double)i;
  return r;
}

__device__ void su2_cg(int j1, int j2, int j3, double cg[5][5][5]) {
  for (int a = 0; a < 5; ++a)
    for (int b = 0; b < 5; ++b)
      for (int c = 0; c < 5; ++c) cg[a][b][c] = 0.0;
  for (int m1 = -j1; m1 <= j1; ++m1)
    for (int m2 = -j2; m2 <= j2; ++m2) {
      int m3 = m1 + m2;
      if (m3 < -j3 || m3 > j3) continue;
      double pref = sqrt((double)(2 * j3 + 1) * dfact(j3 + j1 - j2) * dfact(j3 - j1 + j2) *
                         dfact(j1 + j2 - j3) / dfact(j1 + j2 + j3 + 1));
      pref *= sqrt(dfact(j3 + m3) * dfact(j3 - m3) * dfact(j1 - m1) * dfact(j1 + m1) *
                   dfact(j2 - m2) * dfact(j2 + m2));
      double s = 0.0;
      for (int k = 0; k <= j1 + j2 - j3; ++k) {
        int t0 = k, t1 = j1 + j2 - j3 - k, t2 = j1 - m1 - k;
        int t3 = j2 + m2 - k, t4 = j3 - j2 + m1 + k, t5 = j3 - j1 - m2 + k;
        if (t0 < 0 || t1 < 0 || t2 < 0 || t3 < 0 || t4 < 0 || t5 < 0) continue;
        double term = 1.0 / (dfact(t0) * dfact(t1) * dfact(t2) * dfact(t3) * dfact(t4) * dfact(t5));
        s += (k & 1) ? -term : term;
      }
      cg[j1 + m1][j2 + m2][j3 + m3] = pref * s;
    }
  double inv = 1.0 / sqrt((double)(2 * j3 + 1));
  for (int a = 0; a < 5; ++a)
    for (int b = 0; b < 5; ++b)
      for (int c = 0; c < 5; ++c) cg[a][b][c] *= inv;
}

__device__ void make_q(int l, double qr[5][5], double qi[5][5]) {
  for (int a = 0; a < 5; ++a)
    for (int b = 0; b < 5; ++b) { qr[a][b] = 0.0; qi[a][b] = 0.0; }
  const double s = 0.70710678118654752440;
  for (int m = -l; m < 0; ++m) {
    qr[l + m][l - m] = s;   // q[l+m, l+|m|] = 1/sqrt2
    qi[l + m][l + m] = -s;  // q[l+m, l-|m|] = -i/sqrt2
  }
  qr[l][l] = 1.0;
  for (int m = 1; m <= l; ++m) {
    double sgn = (m & 1) ? -1.0 : 1.0;
    qr[l + m][l + m] = sgn * s;  // (-1)^m/sqrt2
    qi[l + m][l - m] = sgn * s;  // i*(-1)^m/sqrt2
  }
  // multiply by (-i)^l
  double pr, pi;
  switch (l & 3) {
    case 0: pr = 1; pi = 0; break;
    case 1: pr = 0; pi = -1; break;
    case 2: pr = -1; pi = 0; break;
    default: pr = 0; pi = 1; break;
  }
  for (int a = 0; a < 5; ++a)
    for (int b = 0; b < 5; ++b) {
      double r = qr[a][b] * pr - qi[a][b] * pi;
      double im = qr[a][b] * pi + qi[a][b] * pr;
      qr[a][b] = r; qi[a][b] = im;
    }
}

// W[ (ig*51 + col)*9 + jg ] = alpha_p * w3j_p[i_local, j_local, k]
__global__ void init_w3j(float* __restrict__ W) {
  int t = threadIdx.x;
  for (int idx = t; idx < 9 * 51 * 9; idx += 32) W[idx] = 0.f;
  __syncthreads();
  if (t >= 15) return;
  int l1 = c_L1[t], l2 = c_L2[t], l3 = c_L3[t];
  double cg[5][5][5];
  su2_cg(l1, l2, l3, cg);
  double q1r[5][5], q1i[5][5], q2r[5][5], q2i[5][5], q3r[5][5], q3i[5][5];
  make_q(l1, q1r, q1i);
  make_q(l2, q2r, q2i);
  make_q(l3, q3r, q3i);
  double alpha = sqrt((double)(2 * l3 + 1));  // fan_in == 1 for every output irrep
  int rb = (l1 == 0) ? 0 : (l1 == 1) ? 1 : 4;
  int jb = (l2 == 0) ? 0 : (l2 == 1) ? 1 : 4;
  int cs = c_colStart[t];
  for (int j = 0; j <= 2 * l1; ++j)
    for (int L = 0; L <= 2 * l2; ++L)
      for (int n = 0; n <= 2 * l3; ++n) {
        double acc = 0.0;
        for (int i = 0; i <= 2 * l1; ++i)
          for (int k = 0; k <= 2 * l2; ++k) {
            int m = i + k - l1 - l2 + l3;  // m3 = m1 + m2 constraint
            if (m < 0 || m > 2 * l3) continue;
            double c = cg[i][k][m];
            if (c == 0.0) continue;
            // t12 = q1[i][j] * q2[k][L];  acc += c * Re(t12 * conj(q3[m][n]))
            double tr = q1r[i][j] * q2r[k][L] - q1i[i][j] * q2i[k][L];
            double ti = q1r[i][j] * q2i[k][L] + q1i[i][j] * q2r[k][L];
            acc += c * (tr * q3r[m][n] + ti * q3i[m][n]);
          }
        W[((rb + j) * 51 + (cs + n)) * 9 + (jb + L)] = (float)(alpha * acc);
      }
}

// ------------------------------ main kernel --------------------------------

__device__ __forceinline__ float x1_at(const float* x1s, int u, int ig) {
  // per-u 9-vector gathered from irrep blocks: l=0 @0 (stride1), l=1 @64 (stride3), l=2 @256 (stride5)
  int addr;
  if (ig == 0)      addr = u;
  else if (ig < 4)  addr = 64 + u * 3 + (ig - 1);
  else              addr = 256 + u * 5 + (ig - 4);
  return (ig < 9) ? x1s[addr] : 0.f;
}

__global__ __launch_bounds__(32) void tp_kernel(const float* __restrict__ x1,
                                                const float* __restrict__ x2,
                                                const float* __restrict__ w,
                                                const float* __restrict__ W,
                                                float* __restrict__ out, int E) {
  int e = blockIdx.x;
  if (e >= E) return;
  int tid = threadIdx.x;

  // s_stage doubles as the x1 staging area (x1 is dead once A-fragments are in
  // registers) and later as the output tile for coalesced 128-bit writeout.
  __shared__ __align__(16) float s_stage[3264];
  __shared__ __align__(16) float wsh[960];
  __shared__ float Gs[12 * 64];  // K padded 9->12, N padded 51->64
  __shared__ float x2s[12];

  float* x1s = s_stage;

  // ---- coalesced 128-bit input staging ----
  const float4* x1e4 = (const float4*)(x1 + (size_t)e * 576);  // 144 float4
  const float4* we4 = (const float4*)(w + (size_t)e * 960);    // 240 float4
  for (int idx = tid; idx < 144; idx += 32) ((float4*)x1s)[idx] = x1e4[idx];
  for (int idx = tid; idx < 240; idx += 32) ((float4*)wsh)[idx] = we4[idx];
  if (tid < 12) x2s[tid] = (tid < 9) ? x2[(size_t)e * 9 + tid] : 0.f;
  for (int idx = tid; idx < 12 * 64; idx += 32) Gs[idx] = 0.f;
  __syncthreads();

  // ---- per-edge G[9x51] = sum_j W[i,col,j] * x2[j] ----
  for (int idx = tid; idx < 9 * 51; idx += 32) {
    int i = idx / 51, c = idx - i * 51;
    const float* wp = W + (i * 51 + c) * 9;
    float acc = 0.f;
#pragma unroll
    for (int j = 0; j < 9; ++j) acc += wp[j] * x2s[j];
    Gs[i * 64 + c] = acc;
  }
  __syncthreads();

  int half = tid >> 4;  // lane group (ISA 16x4 A layout: v0/v1 hold K=2h / K=2h+1)
  int lm = tid & 15;

  // ---- A fragments: A[M=u, K=ig], 4 M-tiles x 3 K-steps ----
  v2f afrag[4][3];
#pragma unroll
  for (int m = 0; m < 4; ++m) {
    int u = 16 * m + lm;
#pragma unroll
    for (int kt = 0; kt < 3; ++kt) {
      int i0 = 4 * kt + 2 * half;
      afrag[m][kt].x = x1_at(x1s, u, i0);
      afrag[m][kt].y = x1_at(x1s, u, i0 + 1);
    }
  }
  __syncthreads();  // x1 region of s_stage is dead; safe to overwrite below

  const v8f zero8 = {0.f, 0.f, 0.f, 0.f, 0.f, 0.f, 0.f, 0.f};

#pragma unroll
  for (int nt = 0; nt < 4; ++nt) {
    v8f acc[4] = {zero8, zero8, zero8, zero8};
#pragma unroll
    for (int kt = 0; kt < 3; ++kt) {
      int kr = 4 * kt + 2 * half;
      v2f b;
      b.x = Gs[kr * 64 + 16 * nt + lm];
      b.y = Gs[(kr + 1) * 64 + 16 * nt + lm];
#pragma unroll
      for (int m = 0; m < 4; ++m) {
        acc[m] = __builtin_amdgcn_wmma_f32_16x16x4_f32(
            /*neg_a=*/false, afrag[m][kt], /*neg_b=*/false, b,
            /*c_mod=*/(short)0, acc[m], /*reuse_a=*/false, /*reuse_b=*/false);
      }
    }
    // Scale by per-(u,path) weight and scatter into the LDS staging tile
    // (sorted e3nn output layout), using cheap 32-bit ds addressing.
    int c = 16 * nt + lm;
    bool valid = (c < 51);
    int cc = valid ? c : 0;
    int p = c_col2p[cc];
    int k = c_col2k[cc];
    int dimc = c_pdim[p];
    int obase = c_outOff[p] + k;
    int wbase = 64 * p;
#pragma unroll
    for (int m = 0; m < 4; ++m) {
#pragma unroll
      for (int v = 0; v < 8; ++v) {
        int u = 16 * m + v + 8 * half;  // C/D layout: M = v + 8*half per 16x16 tile
        float val = acc[m][v] * wsh[wbase + u];
        if (valid) s_stage[obase + u * dimc] = val;
      }
    }
  }
  __syncthreads();

  // ---- coalesced 128-bit writeout: 3264 floats = 816 float4 ----
  float4* oute4 = (float4*)(out + (size_t)e * 3264);
  const float4* st4 = (const float4*)s_stage;
  for (int idx = tid; idx < 816; idx += 32) oute4[idx] = st4[idx];
}

// ------------------------------- launcher ----------------------------------

extern "C" void kernel_launch(void* const* d_in, const int* in_sizes, int n_in,
                              void* d_out, int out_size, void* d_ws, size_t ws_size,
                              hipStream_t stream) {
  (void)n_in; (void)out_size; (void)ws_size;
  const float* x1 = (const float*)d_in[0];  // [E, 576]
  const float* x2 = (const float*)d_in[1];  // [E, 9]
  const float* w = (const float*)d_in[2];   // [E, 960]
  float* out = (float*)d_out;               // [E, 3264]
  float* W = (float*)d_ws;                  // 9*51*9 floats (alpha-folded w3j)
  int E = in_sizes[0] / 576;

  hipLaunchKernelGGL(init_w3j, dim3(1), dim3(32), 0, stream, W);
  hipLaunchKernelGGL(tp_kernel, dim3(E), dim3(32), 0, stream, x1, x2, w, W, out, E);
}